// InformerEncoder_57166014710077
// MI455X (gfx1250) — compile-verified
//
#include <hip/hip_runtime.h>
#include <math.h>

typedef __attribute__((ext_vector_type(2))) float v2f;
typedef __attribute__((ext_vector_type(8))) float v8f;
typedef int v4i_t __attribute__((__vector_size__(16)));

#define Hh 8
#define Cc 256
#define Bf 32

#if __has_builtin(__builtin_amdgcn_global_load_async_to_lds_b128)
#define USE_ASYNC_LDS 1
#endif
#if __has_builtin(__builtin_amdgcn_s_wait_asynccnt)
#define WAIT_ASYNC() __builtin_amdgcn_s_wait_asynccnt(0)
#else
#define WAIT_ASYNC() asm volatile("s_wait_asynccnt 0x0" ::: "memory")
#endif

// ---------------------------------------------------------------------------
// Generic WMMA GEMM: Y(M x 256) = X(M x K) @ W(K x 256) + bias
//   mode_a  : 0 = plain row-major X (row stride K)
//             1 = causal width-2 conv concat (K=512):
//                 A[m][k] = k<256 ? X[m-1][k] (0 at l==0) : X[m][k-256]
//   mode_out: 0 = row-major Y[m*256+n]
//             1 = q/k/v layout Y[((bf*H+h)*L+l)*32+d], col n -> (h,d)
// 256 threads = 8 wave32's; wave w owns output columns [32w, 32w+32)
// (two 16x16 D tiles sharing one A fragment -> 2 WMMAs per A read pair).
// A tile (16 rows x K) staged in LDS via GLOBAL_LOAD_ASYNC_TO_LDS_B128 when
// available; B streamed from global (weights L2-hot) with global_prefetch.
// fp32 end-to-end via V_WMMA_F32_16X16X4_F32 (workload is bandwidth-bound at
// ~100 GFLOP over an L2-resident 64MB working set, so full precision is free).
// ---------------------------------------------------------------------------
__global__ __launch_bounds__(256)
void wmma_gemm_kernel(const float* __restrict__ X, const float* __restrict__ W,
                      const float* __restrict__ bias, float* __restrict__ Y,
                      int L, int K, int mode_a, int mode_out) {
  extern __shared__ float Alds[];           // 16 * K floats
  const int tid  = threadIdx.x;
  const int lane = tid & 31;
  const int wave = tid >> 5;                // 0..7
  const int n0   = wave * 32;
  const int m0   = blockIdx.x * 16;

#ifdef USE_ASYNC_LDS
  const bool can_async = (mode_a == 0) || ((m0 % L) != 0);
  if (can_async) {
    const int nvec = (16 * K) >> 2;         // 128-bit transfers
    for (int i4 = tid; i4 < nvec; i4 += 256) {
      const float* src;
      if (mode_a == 0) {
        src = X + (size_t)m0 * K + (size_t)i4 * 4;
      } else {                              // K == 512 concat, no pad needed here
        int r = i4 >> 7, k = (i4 & 127) << 2;
        int m = m0 + r;
        src = (k < 256) ? (X + (size_t)(m - 1) * 256 + k)
                        : (X + (size_t)m * 256 + (k - 256));
      }
      __builtin_amdgcn_global_load_async_to_lds_b128(
          (__attribute__((address_space(1))) v4i_t*)(float*)src,
          (__attribute__((address_space(3))) v4i_t*)(Alds + (size_t)i4 * 4),
          0, 0);
    }
    WAIT_ASYNC();
  } else
#endif
  {
    if (mode_a == 0) {
      for (int idx = tid; idx < 16 * K; idx += 256) {
        int r = idx / K, kk = idx % K;
        Alds[idx] = X[(size_t)(m0 + r) * K + kk];
      }
    } else {                                // sequence-start tile: zero left pad
      for (int idx = tid; idx < 16 * 512; idx += 256) {
        int r = idx >> 9, kk = idx & 511;
        int m = m0 + r;
        int l = m % L;
        float val;
        if (kk < 256) val = (l == 0) ? 0.0f : X[(size_t)(m - 1) * 256 + kk];
        else          val = X[(size_t)m * 256 + (kk - 256)];
        Alds[idx] = val;
      }
    }
  }
  __syncthreads();

  const int nlane = lane & 15;              // column within a 16-wide tile
  const int mlane = lane & 15;              // A row (both lane halves M=0..15)
  const int kb    = (lane >> 4) * 2;        // K-pair select per ISA A/B layout

  v8f acc0 = {}, acc1 = {};
  const float* Wp0 = W + n0 + nlane;
  const float* Wp1 = Wp0 + 16;
  #pragma unroll 4
  for (int kk = 0; kk < K; kk += 4) {
    if ((kk & 63) == 0 && (kk + 64) < K) {  // prefetch next B K-chunk (L2->WGP$)
      __builtin_prefetch(Wp0 + (size_t)(kk + 64) * 256, 0, 1);
      __builtin_prefetch(Wp1 + (size_t)(kk + 64) * 256, 0, 1);
    }
    v2f a, b0, b1;
    a.x  = Alds[mlane * K + kk + kb];
    a.y  = Alds[mlane * K + kk + kb + 1];
    b0.x = Wp0[(size_t)(kk + kb) * 256];
    b0.y = Wp0[(size_t)(kk + kb + 1) * 256];
    b1.x = Wp1[(size_t)(kk + kb) * 256];
    b1.y = Wp1[(size_t)(kk + kb + 1) * 256];
    acc0 = __builtin_amdgcn_wmma_f32_16x16x4_f32(false, a, false, b0,
                                                 (short)0, acc0, false, false);
    acc1 = __builtin_amdgcn_wmma_f32_16x16x4_f32(false, a, false, b1,
                                                 (short)0, acc1, false, false);
  }

  const float bn0 = bias[n0 + nlane];
  const float bn1 = bias[n0 + 16 + nlane];
  const int mhi = (lane >> 4) << 3;         // D rows: r + 8*(lane/16)
  if (mode_out == 0) {
    #pragma unroll
    for (int r = 0; r < 8; ++r) {
      int m = m0 + r + mhi;
      Y[(size_t)m * 256 + n0 + nlane]      = acc0[r] + bn0;
      Y[(size_t)m * 256 + n0 + 16 + nlane] = acc1[r] + bn1;
    }
  } else {
    int c0 = n0 + nlane,      h0 = c0 >> 5, d0 = c0 & 31;
    int c1 = n0 + 16 + nlane, h1 = c1 >> 5, d1 = c1 & 31;
    #pragma unroll
    for (int r = 0; r < 8; ++r) {
      int m = m0 + r + mhi;
      int bf = m / L, l = m % L;
      Y[((size_t)(bf * Hh + h0) * L + l) * 32 + d0] = acc0[r] + bn0;
      Y[((size_t)(bf * Hh + h1) * L + l) * 32 + d1] = acc1[r] + bn1;
    }
  }
}

// ---------------------------------------------------------------------------
// Deterministic sampling indices (hash-based stand-in for JAX threefry).
// ---------------------------------------------------------------------------
__global__ void gen_idx_kernel(int* __restrict__ idx, int L, int S, unsigned seed) {
  int i = blockIdx.x * blockDim.x + threadIdx.x;
  if (i >= L * S) return;
  unsigned x = seed ^ (0x9E3779B9u * (unsigned)(i + 1));
  x ^= x >> 16; x *= 0x7FEB352Du; x ^= x >> 15; x *= 0x846CA68Bu; x ^= x >> 16;
  idx[i] = (int)(x % (unsigned)L);
}

// ---------------------------------------------------------------------------
// M(l) = max_s(q_l . k_idx[l,s]) - mean_s(...)   over S sampled keys
// ---------------------------------------------------------------------------
__global__ __launch_bounds__(256)
void mscore_kernel(const float* __restrict__ q, const float* __restrict__ k,
                   const int* __restrict__ idx, float* __restrict__ M,
                   int L, int S) {
  int pair = blockIdx.y;                       // bf*H + h
  int l = blockIdx.x * 256 + threadIdx.x;
  if (l >= L) return;
  const float4* qv = (const float4*)(q + ((size_t)pair * L + l) * 32);
  float4 qr[8];
  #pragma unroll
  for (int j = 0; j < 8; ++j) qr[j] = qv[j];
  float mx = -3.0e38f, sm = 0.0f;
  for (int s = 0; s < S; ++s) {
    int ks = idx[l * S + s];
    const float4* kv = (const float4*)(k + ((size_t)pair * L + ks) * 32);
    float d = 0.0f;
    #pragma unroll
    for (int j = 0; j < 8; ++j) {
      float4 kk = kv[j];
      d += qr[j].x * kk.x + qr[j].y * kk.y + qr[j].z * kk.z + qr[j].w * kk.w;
    }
    mx = fmaxf(mx, d); sm += d;
  }
  M[(size_t)pair * L + l] = mx - sm / (float)S;
}

// ---------------------------------------------------------------------------
// Top-NT selection per (bf,h): one wave32, LDS-resident scores, shfl argmax.
// ---------------------------------------------------------------------------
__global__ __launch_bounds__(32)
void topk_kernel(const float* __restrict__ M, int* __restrict__ mtop,
                 int L, int NT) {
  __shared__ float sm[2048];
  int pair = blockIdx.x, lane = threadIdx.x;
  for (int i = lane; i < L; i += 32) sm[i] = M[(size_t)pair * L + i];
  __syncthreads();
  for (int t = 0; t < NT; ++t) {
    float bv = -3.4e38f; int bi = 0x7FFFFFFF;
    for (int i = lane; i < L; i += 32) {
      float v = sm[i];
      if (v > bv) { bv = v; bi = i; }
    }
    #pragma unroll
    for (int off = 16; off > 0; off >>= 1) {
      float ov = __shfl_xor(bv, off);
      int   oi = __shfl_xor(bi, off);
      if (ov > bv || (ov == bv && oi < bi)) { bv = ov; bi = oi; }
    }
    if (lane == 0) { mtop[pair * NT + t] = bi; sm[bi] = -3.4e38f; }
    __syncthreads();
  }
}

// ---------------------------------------------------------------------------
// attnin[bf, l, h*32+d] = mean_l(v[bf,h,:,d])  broadcast to all rows
// ---------------------------------------------------------------------------
__global__ __launch_bounds__(256)
void vmean_bcast_kernel(const float* __restrict__ v, float* __restrict__ attnin,
                        int L) {
  __shared__ float part[8][32];
  __shared__ float vbar[32];
  int pair = blockIdx.x, bf = pair >> 3, h = pair & 7;
  int d = threadIdx.x & 31, g = threadIdx.x >> 5;
  float s = 0.0f;
  for (int l = g; l < L; l += 8) s += v[((size_t)pair * L + l) * 32 + d];
  part[g][d] = s;
  __syncthreads();
  if (threadIdx.x < 32) {
    float t = 0.0f;
    #pragma unroll
    for (int j = 0; j < 8; ++j) t += part[j][threadIdx.x];
    vbar[threadIdx.x] = t / (float)L;
  }
  __syncthreads();
  for (int i = threadIdx.x; i < L * 32; i += 256) {
    int l = i >> 5, dd = i & 31;
    attnin[((size_t)bf * L + l) * 256 + h * 32 + dd] = vbar[dd];
  }
}

// ---------------------------------------------------------------------------
// Dense attention for one selected query row; deterministic LDS reductions.
// Scatters result into attnin (overwrites mean row).
// ---------------------------------------------------------------------------
__global__ __launch_bounds__(256)
void attn_kernel(const float* __restrict__ q, const float* __restrict__ k,
                 const float* __restrict__ v, const int* __restrict__ mtop,
                 float* __restrict__ attnin, int L, int NT) {
  __shared__ float sc[2048];
  __shared__ float qr[32];
  __shared__ float red[256];
  __shared__ float part[8][32];
  int u = blockIdx.x % NT, pair = blockIdx.x / NT;
  int bf = pair >> 3, h = pair & 7;
  int tid = threadIdx.x;
  int lq = mtop[pair * NT + u];
  if (tid < 32) qr[tid] = q[((size_t)pair * L + lq) * 32 + tid];
  __syncthreads();

  const float scale = 0.17677669529663687f;   // 1/sqrt(32)
  float lmax = -3.0e38f;
  for (int l = tid; l < L; l += 256) {
    const float4* kv = (const float4*)(k + ((size_t)pair * L + l) * 32);
    const float4* qv = (const float4*)qr;
    float d = 0.0f;
    #pragma unroll
    for (int j = 0; j < 8; ++j) {
      float4 kk = kv[j], qq = qv[j];
      d += qq.x * kk.x + qq.y * kk.y + qq.z * kk.z + qq.w * kk.w;
    }
    d *= scale;
    sc[l] = d;
    lmax = fmaxf(lmax, d);
  }
  red[tid] = lmax; __syncthreads();
  for (int off = 128; off; off >>= 1) {
    if (tid < off) red[tid] = fmaxf(red[tid], red[tid + off]);
    __syncthreads();
  }
  float mx = red[0];
  __syncthreads();

  float lsum = 0.0f;
  for (int l = tid; l < L; l += 256) {
    float e = __expf(sc[l] - mx);
    sc[l] = e;
    lsum += e;
  }
  red[tid] = lsum; __syncthreads();
  for (int off = 128; off; off >>= 1) {
    if (tid < off) red[tid] += red[tid + off];
    __syncthreads();
  }
  float ssum = red[0];
  __syncthreads();

  // upd[d] = sum_l attn[l] * v[l][d], deterministic 8-way partial reduce
  int d = tid & 31, g = tid >> 5;
  float a = 0.0f;
  for (int l = g; l < L; l += 8)
    a += sc[l] * v[((size_t)pair * L + l) * 32 + d];
  part[g][d] = a;
  __syncthreads();
  if (tid < 32) {
    float s2 = 0.0f;
    #pragma unroll
    for (int j = 0; j < 8; ++j) s2 += part[j][tid];
    attnin[((size_t)bf * L + lq) * 256 + h * 32 + tid] = s2 / ssum;
  }
}

// ---------------------------------------------------------------------------
// ELU -> maxpool(2 along L) -> optional LayerNorm(C=256)
// ---------------------------------------------------------------------------
__global__ __launch_bounds__(256)
void pool_ln_kernel(const float* __restrict__ y, const float* __restrict__ g,
                    const float* __restrict__ be, float* __restrict__ out,
                    int Lout, int do_ln) {
  __shared__ float red[256];
  int row = blockIdx.x;
  int c = threadIdx.x;
  int bf = row / Lout, l2 = row % Lout;
  size_t base = ((size_t)bf * (2 * Lout) + 2 * l2) * 256;
  float y0 = y[base + c], y1 = y[base + 256 + c];
  float e0 = y0 > 0.0f ? y0 : __expf(y0) - 1.0f;
  float e1 = y1 > 0.0f ? y1 : __expf(y1) - 1.0f;
  float m = fmaxf(e0, e1);
  if (do_ln) {
    red[c] = m; __syncthreads();
    for (int off = 128; off; off >>= 1) {
      if (c < off) red[c] += red[c + off];
      __syncthreads();
    }
    float mu = red[0] * (1.0f / 256.0f);
    __syncthreads();
    float dv = (m - mu) * (m - mu);
    red[c] = dv; __syncthreads();
    for (int off = 128; off; off >>= 1) {
      if (c < off) red[c] += red[c + off];
      __syncthreads();
    }
    float var = red[0] * (1.0f / 256.0f);
    out[(size_t)row * 256 + c] = g[c] * (m - mu) * rsqrtf(var + 1.0e-3f) + be[c];
  } else {
    out[(size_t)row * 256 + c] = m;
  }
}

// ---------------------------------------------------------------------------
extern "C" void kernel_launch(void* const* d_in, const int* in_sizes, int n_in,
                              void* d_out, int out_size, void* d_ws, size_t ws_size,
                              hipStream_t stream) {
  (void)in_sizes; (void)n_in; (void)out_size; (void)ws_size;
  const float* x = (const float*)d_in[0];

  // params flattened in dict order: per layer Wq,bq,Wk,bk,Wv,bv,Wo,bo,Wc,bc[,g,be]
  const int layer_base[3] = {1, 1 + 12, 1 + 24};

  float* ws = (float*)d_ws;
  const size_t SEG = (size_t)Bf * 2048 * 256;     // 16M floats = 64MB
  float* T0 = ws;                  // next-layer activations
  float* Q  = ws + 1 * SEG;        // also reused as A (post-Wo) buffer
  float* Kb = ws + 2 * SEG;
  float* V  = ws + 3 * SEG;
  float* Xb = ws + 4 * SEG;        // attnin, then conv output
  float* Ab = Q;                   // Wo output overwrites Q (dead after attn)
  int*   idxb  = (int*)(ws + 5 * SEG);
  float* Mb    = (float*)(idxb + 2048 * 64);
  int*   mtopb = (int*)(Mb + (size_t)Bf * Hh * 2048);

  const float* cur = x;
  int L = 2048;
  for (int layer = 0; layer < 3; ++layer) {
    const int pb = layer_base[layer];
    const float* Wq = (const float*)d_in[pb + 0];
    const float* bq = (const float*)d_in[pb + 1];
    const float* Wk = (const float*)d_in[pb + 2];
    const float* bk = (const float*)d_in[pb + 3];
    const float* Wv = (const float*)d_in[pb + 4];
    const float* bv = (const float*)d_in[pb + 5];
    const float* Wo = (const float*)d_in[pb + 6];
    const float* bo = (const float*)d_in[pb + 7];
    const float* Wc = (const float*)d_in[pb + 8];   // (2,256,256) == (512,256)
    const float* bc = (const float*)d_in[pb + 9];
    const float* gg = (layer < 2) ? (const float*)d_in[pb + 10] : nullptr;
    const float* be = (layer < 2) ? (const float*)d_in[pb + 11] : nullptr;

    const int NT = (layer == 0) ? 40 : 35;          // 5*ceil(ln L)
    const int Mrows = Bf * L;
    const size_t lds256 = 16 * 256 * sizeof(float);
    const size_t lds512 = 16 * 512 * sizeof(float);

    // Q/K/V projections (WMMA GEMM, qkv output layout)
    wmma_gemm_kernel<<<Mrows / 16, 256, lds256, stream>>>(cur, Wq, bq, Q,  L, 256, 0, 1);
    wmma_gemm_kernel<<<Mrows / 16, 256, lds256, stream>>>(cur, Wk, bk, Kb, L, 256, 0, 1);
    wmma_gemm_kernel<<<Mrows / 16, 256, lds256, stream>>>(cur, Wv, bv, V,  L, 256, 0, 1);

    // ProbSparse scoring + selection
    gen_idx_kernel<<<(L * NT + 255) / 256, 256, 0, stream>>>(idxb, L, NT, 0xC0FFEEu + layer);
    mscore_kernel<<<dim3(L / 256, Bf * Hh), 256, 0, stream>>>(Q, Kb, idxb, Mb, L, NT);
    topk_kernel<<<Bf * Hh, 32, 0, stream>>>(Mb, mtopb, L, NT);

    // ctx = broadcast v-mean, then scatter dense attention rows
    vmean_bcast_kernel<<<Bf * Hh, 256, 0, stream>>>(V, Xb, L);
    attn_kernel<<<Bf * Hh * NT, 256, 0, stream>>>(Q, Kb, V, mtopb, Xb, L, NT);

    // output projection (Q now dead -> reuse as Ab)
    wmma_gemm_kernel<<<Mrows / 16, 256, lds256, stream>>>(Xb, Wo, bo, Ab, L, 256, 0, 0);

    // causal width-2 conv as K=512 WMMA GEMM
    wmma_gemm_kernel<<<Mrows / 16, 256, lds512, stream>>>(Ab, Wc, bc, Xb, L, 512, 1, 0);

    // ELU + maxpool(2) + LN (layers 0,1) / final output (layer 2)
    const int Lout = L / 2;
    if (layer < 2) {
      pool_ln_kernel<<<Bf * Lout, 256, 0, stream>>>(Xb, gg, be, T0, Lout, 1);
      cur = T0;
      L = Lout;
    } else {
      pool_ln_kernel<<<Bf * Lout, 256, 0, stream>>>(Xb, nullptr, nullptr,
                                                    (float*)d_out, Lout, 0);
    }
  }
}